// CppnPotentialCA_27539330302220
// MI455X (gfx1250) — compile-verified
//
#include <hip/hip_runtime.h>
#include <math.h>

// ---------------------------------------------------------------------------
// CppnPotentialCA on MI455X (gfx1250)
//
// Phase 1 (ca_conv_wmma): grouped 15^3 conv via V_WMMA_F32_16X16X4_F32.
//   M = 16 x-positions, N = 16 (6 kernels of one input channel), K = taps.
//   Each wave batches its 4 z-rows against one B fragment (4 WMMAs / B load).
//   Kernel weights are DMA'd by the Tensor Data Mover as a 15-wide tensor
//   read with a 16-wide tile: the OOB element zero-fills tap 15 in LDS, so
//   the B operand needs no masking in the hot loop.
// Phase 2 (ca_alive): 3x3x3 wrap max-pool "alive" gate, final output.
// ---------------------------------------------------------------------------

typedef __attribute__((ext_vector_type(2))) float v2f;
typedef __attribute__((ext_vector_type(8))) float v8f;
typedef __attribute__((ext_vector_type(4))) unsigned int u32x4;
typedef __attribute__((ext_vector_type(8))) int i32x8;
typedef __attribute__((ext_vector_type(4))) int i32x4;

#define ZDIM 128
#define CDIM 6
#define RR 7
#define DD 15              // kernel diameter
#define DDP 16             // zero-padded tap row in LDS
#define KVOL (DD*DD*DD)    // 3375 taps
#define NROWS (CDIM*DD*DD) // 1350 tap-rows per channel group
#define TX 16
#define TY 8
#define TZ 4
#define HX (TX + 2*RR)     // 30
#define HXP 32             // padded LDS row (covers the padded 16th-tap read)
#define HY (TY + 2*RR)     // 22
#define HZ (TZ + 2*RR)     // 18
#define ZSTRIDE (HY*HXP)   // float stride between z-planes in xs
#define PLANE ((size_t)ZDIM*ZDIM*ZDIM) // 2097152 elements per channel

// --- TDM: DMA `rows` rows of `rowlen` f32 from global into LDS, writing ----
// --- `rowpad` elements per row; elements beyond rowlen are OOB -> zero. ----
__device__ __forceinline__ void tdm_load_rows_padded(const float* gsrc,
                                                     float* ldst, int rowlen,
                                                     int rowpad, int rows) {
  // D# group 0: count=1 | lds_addr | global_addr | type=2
  u32x4 g0;
  g0.x = 1u;                                             // count = 1 valid D#
  g0.y = (unsigned)(unsigned long long)(void*)ldst;      // LDS byte offset
  unsigned long long ga = (unsigned long long)gsrc;
  g0.z = (unsigned)ga;
  g0.w = (unsigned)((ga >> 32) & 0x01FFFFFFull) | (2u << 30); // addr hi|type=2
  // D# group 1: data_size=4B, 2-D tile rows x rowpad over rows x rowlen tensor
  i32x8 g1;
  g1[0] = (2 << 16);                                     // data_size = 4 bytes
  g1[1] = (int)(((unsigned)rowlen & 0xFFFFu) << 16);     // tensor_dim0[15:0]
  g1[2] = (int)(((unsigned)rowlen >> 16) |
                (((unsigned)rows & 0xFFFFu) << 16));     // dim0 hi | dim1 lo
  g1[3] = (int)(((unsigned)rows >> 16) |
                (((unsigned)rowpad & 0xFFFFu) << 16));   // dim1 hi | tile_dim0
  g1[4] = rows;                                          // tile_dim1
  g1[5] = rowlen;                                        // tensor_dim0_stride
  g1[6] = 0;
  g1[7] = 0;
  i32x4 z4 = {0, 0, 0, 0};
  i32x8 z8 = {0, 0, 0, 0, 0, 0, 0, 0};
  __builtin_amdgcn_tensor_load_to_lds(g0, g1, z4, z4, z8, 0);
}

__global__ __launch_bounds__(256, 1)
void ca_conv_wmma(const float* __restrict__ x, const float* __restrict__ kern,
                  const float* __restrict__ m, const float* __restrict__ s,
                  const float* __restrict__ h, const float* __restrict__ T,
                  const float* __restrict__ rnd, float* __restrict__ pre) {
  __shared__ float xs[HZ][HY][HXP];        // 50.7 KB halo tile (one channel)
  __shared__ float klds[NROWS * DDP + 8];  // 86.4 KB zero-padded kernel rows
  __shared__ float fld[TZ * TY][TX][8];    // 16 KB field accumulators

  const int tid  = threadIdx.x;
  const int lane = tid & 31;
  const int wave = tid >> 5;
  const int x0 = blockIdx.x * TX;
  const int y0 = blockIdx.y * TY;
  const int z0 = blockIdx.z * TZ;

  // zero field accumulators
  for (int i = tid; i < TZ * TY * TX * 8; i += 256) (&fld[0][0][0])[i] = 0.f;

  const int  j   = lane & 15;        // N column (kernel id) == M row (x pos)
  const int  kb  = (lane >> 4) * 2;  // per-lane K base within the 4-chunk
  const bool jv  = (j < CDIM);
  const int  jj  = jv ? j : (CDIM - 1);   // clamped kernel slot for B reads
  const int  xoff = j + kb;               // per-lane A base offset
  const int  yy   = wave;                 // this wave's y-row in the tile

  for (int c0 = 0; c0 < CDIM; ++c0) {
    __syncthreads();  // previous iteration done reading xs/klds

    // --- TDM: DMA + zero-pad the 6 kernels of channel c0 into LDS ---------
    if (tid < 32) {
      tdm_load_rows_padded(kern + (size_t)c0 * CDIM * KVOL, klds,
                           DD, DDP, NROWS);
      __builtin_amdgcn_s_wait_tensorcnt(0);
    }

    // --- cooperative wrap-padded halo load of channel c0 ------------------
    const float* xc = x + (size_t)c0 * PLANE;
    if (c0 + 1 < CDIM) __builtin_prefetch(x + (size_t)(c0 + 1) * PLANE, 0, 0);
    for (int i = tid; i < HZ * HY * HX; i += 256) {
      int lz = i / (HY * HX);
      int r2 = i - lz * (HY * HX);
      int ly = r2 / HX;
      int lx = r2 - ly * HX;
      int gz = (z0 - RR + lz) & 127;
      int gy = (y0 - RR + ly) & 127;
      int gx = (x0 - RR + lx) & 127;
      xs[lz][ly][lx] = xc[(size_t)(((gz << 7) + gy) << 7) + gx];
    }
    __syncthreads();  // klds (TDM) + xs visible to all waves

    // per-lane growth parameters for kernel n = c0*6 + j
    const int n = c0 * CDIM + j;
    const float mn = jv ? m[n] : 0.f;
    const float sn = jv ? s[n] : 1.f;
    const float hn = jv ? h[n] : 0.f;
    const float neg_inv2s2 = -1.f / (2.f * sn * sn);

    // per-lane B base: kernel slot jj, K base kb (8B aligned -> b64 loads)
    const float* kbp = &klds[jj * (DD * DD * DDP) + kb];

    // 4 rows per wave: (zz = 0..3, yy = wave), shared B fragment
    v8f acc0 = {}, acc1 = {}, acc2 = {}, acc3 = {};
#pragma unroll 1
    for (int dz = 0; dz < DD; ++dz) {
#pragma unroll 1
      for (int dy = 0; dy < DD; ++dy) {
        const float* xr = &xs[dz][yy + dy][xoff];
        const float* kr = kbp + (dz * DD + dy) * DDP;
#pragma unroll
        for (int dxb = 0; dxb < 4; ++dxb) {
          const int dx0 = dxb * 4;
          // B (4x16 f32): unconditional aligned pair (tap 15 is TDM zero)
          const v2f b = *(const v2f*)(kr + dx0);
          // A (16x4 f32): lanes 0-15 K={0,1}, lanes 16-31 K={2,3}
          v2f a0, a1, a2, a3;
          a0.x = xr[dx0];                a0.y = xr[dx0 + 1];
          a1.x = xr[dx0 + ZSTRIDE];      a1.y = xr[dx0 + ZSTRIDE + 1];
          a2.x = xr[dx0 + 2 * ZSTRIDE];  a2.y = xr[dx0 + 2 * ZSTRIDE + 1];
          a3.x = xr[dx0 + 3 * ZSTRIDE];  a3.y = xr[dx0 + 3 * ZSTRIDE + 1];
          acc0 = __builtin_amdgcn_wmma_f32_16x16x4_f32(
              false, a0, false, b, (short)0, acc0, false, false);
          acc1 = __builtin_amdgcn_wmma_f32_16x16x4_f32(
              false, a1, false, b, (short)0, acc1, false, false);
          acc2 = __builtin_amdgcn_wmma_f32_16x16x4_f32(
              false, a2, false, b, (short)0, acc2, false, false);
          acc3 = __builtin_amdgcn_wmma_f32_16x16x4_f32(
              false, a3, false, b, (short)0, acc3, false, false);
        }
      }
    }

    // C layout: lane l, VGPR v -> M = v + 8*(l>=16), N = l&15.
    // Each (row, x, c1) owned by exactly one (lane, v): plain RMW into LDS.
    if (jv) {
      const int xbase = (lane >> 4) << 3;
#pragma unroll
      for (int zz = 0; zz < TZ; ++zz) {
        const v8f acc = (zz == 0) ? acc0 : (zz == 1) ? acc1
                        : (zz == 2) ? acc2 : acc3;
        const int r = zz * TY + yy;
#pragma unroll
        for (int v = 0; v < 8; ++v) {
          const float d = acc[v] - mn;
          const float u = __expf(d * d * neg_inv2s2) * 2.f - 1.f;
          fld[r][xbase + v][j] += hn * u;
        }
      }
    }
  }
  __syncthreads();

  // --- masked update, clip, write pre-activation field to workspace -------
  const float invT = 1.f / T[0];
  for (int p = tid; p < TZ * TY * TX; p += 256) {
    const int lz = p / (TY * TX);
    const int r2 = p - lz * (TY * TX);
    const int ly = r2 / TX;
    const int lx = r2 - ly * TX;
    const size_t base =
        (size_t)((((z0 + lz) << 7) + (y0 + ly)) << 7) + (x0 + lx);
#pragma unroll 1
    for (int c1 = 0; c1 < CDIM; ++c1) {
      const size_t g = (size_t)c1 * PLANE + base;
      const float f = fld[lz * TY + ly][lx][c1];
      const float mask = floorf(rnd[g] + 0.5f);
      float val = x[g] + mask * f * invT;
      val = fminf(fmaxf(val, 0.f), 1.f);
      pre[g] = val;
    }
  }
}

// --- Phase 2: 3x3x3 wrap max-pool alive gate --------------------------------
__global__ __launch_bounds__(256, 1)
void ca_alive(const float* __restrict__ pre, const float* __restrict__ x,
              float* __restrict__ out, int total) {
  int idx = blockIdx.x * 256 + threadIdx.x;
  if (idx >= total) return;
  const float thr = x[0];  // x[0,0,0,0,0]
  const int xq = idx & 127;
  int t = idx >> 7;
  const int yq = t & 127;
  t >>= 7;
  const int zq = t & 127;
  const int c = t >> 7;
  const float* pc = pre + (size_t)c * PLANE;
  float mx = -INFINITY;
#pragma unroll
  for (int dz = -1; dz <= 1; ++dz)
#pragma unroll
    for (int dy = -1; dy <= 1; ++dy)
#pragma unroll
      for (int dx = -1; dx <= 1; ++dx) {
        const int gz = (zq + dz) & 127, gy = (yq + dy) & 127,
                  gx = (xq + dx) & 127;
        mx = fmaxf(mx, pc[(size_t)(((gz << 7) + gy) << 7) + gx]);
      }
  const float v = pc[(size_t)(((zq << 7) + yq) << 7) + xq];
  out[idx] = (mx >= thr) ? v : 0.f;
}

extern "C" void kernel_launch(void* const* d_in, const int* in_sizes, int n_in,
                              void* d_out, int out_size, void* d_ws,
                              size_t ws_size, hipStream_t stream) {
  const float* x    = (const float*)d_in[0];
  const float* kern = (const float*)d_in[1];
  const float* m    = (const float*)d_in[2];
  const float* s    = (const float*)d_in[3];
  const float* h    = (const float*)d_in[4];
  const float* T    = (const float*)d_in[5];
  const float* rnd  = (const float*)d_in[6];
  // c0_idx / c1_idx (d_in[7], d_in[8]) are n/6 and n%6 by construction.

  float* pre = (float*)d_ws;                 // 6*128^3 f32 = 50.3 MB scratch
  dim3 grid(ZDIM / TX, ZDIM / TY, ZDIM / TZ);
  ca_conv_wmma<<<grid, 256, 0, stream>>>(x, kern, m, s, h, T, rnd, pre);

  const int total = CDIM * (int)PLANE;
  ca_alive<<<(total + 255) / 256, 256, 0, stream>>>(pre, x, (float*)d_out,
                                                    total);
}